// Classifier_61959198212564
// MI455X (gfx1250) — compile-verified
//
#include <hip/hip_runtime.h>
#include <hip/hip_bf16.h>

typedef float v2f __attribute__((ext_vector_type(2)));
typedef float v8f __attribute__((ext_vector_type(8)));

#define NNODES 50000
#define NEDGES 800000
#define NEG_SLOPE 0.2f

// ---------------------------------------------------------------------------
// Staging: pad X rows [rows,K] -> [rows,Kpad] (zero fill)
// ---------------------------------------------------------------------------
__global__ void pad_rows(const float* __restrict__ src, float* __restrict__ dst,
                         int K, int Kpad, long total) {
    const long idx = (long)blockIdx.x * blockDim.x + threadIdx.x;
    if (idx >= total) return;
    const long r = idx / Kpad;
    const int  k = (int)(idx % Kpad);
    dst[idx] = (k < K) ? src[r * K + k] : 0.0f;
}

// Staging: W[K,HD] -> transposed, padded Wt[HDpad, Kpad] (zero fill).
// Makes each lane's B operand a contiguous 8-byte pair in the GEMM loop.
__global__ void stage_wt(const float* __restrict__ W, float* __restrict__ Wt,
                         int K, int HD, int Kpad, int HDpad) {
    const int idx = blockIdx.x * blockDim.x + threadIdx.x;
    if (idx >= HDpad * Kpad) return;
    const int c = idx / Kpad;
    const int k = idx % Kpad;
    Wt[idx] = (c < HD && k < K) ? W[(long)k * HD + c] : 0.0f;
}

// ---------------------------------------------------------------------------
// GEMM: Hout[N, HD] = Xp[N, Kpad] @ Wt[HDpad, Kpad]^T via V_WMMA_F32_16X16X4_F32.
// 256 threads = 8 waves; wave w computes M-tile (blockIdx.x*8 + w), N-tile
// blockIdx.y. Guard-free inner loop: two aligned b64 loads + one WMMA.
// A layout (32-bit 16x4): lane l holds row (l&15), K pair 2*(l>>4)+{0,1}
// C layout: VGPR r -> row r + 8*(l>>4), col (l&15)
// ---------------------------------------------------------------------------
__global__ __launch_bounds__(256)
void gemm16x16x4(const float* __restrict__ Xp, const float* __restrict__ Wt,
                 float* __restrict__ Hout, int Kpad, int HD) {
    const int wave = threadIdx.x >> 5;
    const int mt   = blockIdx.x * 8 + wave;
    if (mt >= NNODES / 16) return;           // wave-uniform: EXEC stays all-ones
    const int nt   = blockIdx.y;
    const int lane = threadIdx.x & 31;
    const int half = lane >> 4;
    const int l15  = lane & 15;
    const int row  = mt * 16 + l15;
    const int col  = nt * 16 + l15;          // always < HDpad (B is padded)

    const float* ap = Xp + (size_t)row * Kpad + half * 2;
    const float* bp = Wt + (size_t)col * Kpad + half * 2;

    v8f c = {};
    const int ktiles = Kpad >> 2;
    #pragma unroll 4
    for (int kt = 0; kt < ktiles; ++kt) {
        const v2f a = *(const v2f*)ap;
        const v2f b = *(const v2f*)bp;
        ap += 4;
        bp += 4;
        c = __builtin_amdgcn_wmma_f32_16x16x4_f32(
                /*neg_a=*/false, a, /*neg_b=*/false, b,
                /*c_mod=*/(short)0, c, /*reuse_a=*/false, /*reuse_b=*/false);
    }

    if (col < HD) {
        #pragma unroll
        for (int r = 0; r < 8; ++r) {
            const int orow = mt * 16 + r + 8 * half;
            Hout[(long)orow * HD + col] = c[r];
        }
    }
}

// el[n,h] = sum_d h[n,h,d]*al[h,d]; er likewise.
__global__ void node_scores(const float* __restrict__ Hout,
                            const float* __restrict__ al,
                            const float* __restrict__ ar,
                            float* __restrict__ el, float* __restrict__ er,
                            int H, int D) {
    const int idx = blockIdx.x * blockDim.x + threadIdx.x;
    if (idx >= NNODES * H) return;
    const int h = idx % H;
    const int n = idx / H;
    const float* hp  = Hout + (long)n * H * D + h * D;
    const float* alp = al + h * D;
    const float* arp = ar + h * D;
    float sl = 0.0f, sr = 0.0f;
    for (int d = 0; d < D; ++d) { const float v = hp[d]; sl += v * alp[d]; sr += v * arp[d]; }
    el[idx] = sl;
    er[idx] = sr;
}

__global__ void init_nodes(unsigned* __restrict__ menc, float* __restrict__ ssum, int total) {
    const int idx = blockIdx.x * blockDim.x + threadIdx.x;
    if (idx >= total) return;
    menc[idx] = 0u;       // below encode(-inf) = 0x007FFFFF -> decodes to "no edge"
    ssum[idx] = 0.0f;
}

__device__ __forceinline__ unsigned enc_ordered(float f) {
    unsigned u = __float_as_uint(f);
    return (u & 0x80000000u) ? ~u : (u | 0x80000000u);
}

// Pass 1: e = leaky_relu(el[src]+er[dst]); atomic max per (dst, head).
__global__ void edge_score_max(const float* __restrict__ el, const float* __restrict__ er,
                               const int* __restrict__ src, const int* __restrict__ dst,
                               float* __restrict__ ebuf, unsigned* __restrict__ menc, int H) {
    const int idx = blockIdx.x * blockDim.x + threadIdx.x;
    if (idx >= NEDGES * H) return;
    const int h = idx % H;
    const int e = idx / H;
    const int s_ = src[e];
    const int d_ = dst[e];
    float v = el[s_ * H + h] + er[d_ * H + h];
    v = (v > 0.0f) ? v : NEG_SLOPE * v;
    ebuf[idx] = v;
    atomicMax(&menc[d_ * H + h], enc_ordered(v));
}

// Decode ordered-uint max; non-finite (isolated dst) -> 0, matching reference.
__global__ void decode_max(const unsigned* __restrict__ menc, float* __restrict__ mf, int total) {
    const int idx = blockIdx.x * blockDim.x + threadIdx.x;
    if (idx >= total) return;
    const unsigned u = menc[idx];
    float f = (u & 0x80000000u) ? __uint_as_float(u ^ 0x80000000u) : __uint_as_float(~u);
    const unsigned raw = __float_as_uint(f);
    if ((raw & 0x7f800000u) == 0x7f800000u) f = 0.0f;   // inf/nan -> 0
    mf[idx] = f;
}

// Pass 2: ex = exp(e - m[dst]); atomic sum per (dst, head).
__global__ void edge_expsum(float* __restrict__ ebuf, const float* __restrict__ mf,
                            const int* __restrict__ dst, float* __restrict__ ssum, int H) {
    const int idx = blockIdx.x * blockDim.x + threadIdx.x;
    if (idx >= NEDGES * H) return;
    const int h = idx % H;
    const int e = idx / H;
    const int d_ = dst[e];
    const float ex = expf(ebuf[idx] - mf[d_ * H + h]);
    ebuf[idx] = ex;
    atomicAdd(&ssum[d_ * H + h], ex);
}

// out[n, j] = bias[j] for j<HD, 0 for pad columns (OS = padded output stride).
__global__ void init_out(float* __restrict__ out, const float* __restrict__ bias,
                         int HD, int OS, int total) {
    const int idx = blockIdx.x * blockDim.x + threadIdx.x;
    if (idx >= total) return;
    const int j = idx % OS;
    out[idx] = (j < HD) ? bias[j] : 0.0f;
}

// Pass 3: out[dst, c] += (ex/sum) * h[src, c], one thread per (edge, feature).
__global__ void edge_aggregate(const float* __restrict__ ebuf, const float* __restrict__ ssum,
                               const float* __restrict__ Hout,
                               const int* __restrict__ src, const int* __restrict__ dst,
                               float* __restrict__ out, int H, int D, int OS) {
    const long idx = (long)blockIdx.x * blockDim.x + threadIdx.x;
    const int HD = H * D;
    const long total = (long)NEDGES * HD;
    if (idx >= total) return;
    const int e = (int)(idx / HD);
    const int c = (int)(idx % HD);
    const int h = c / D;
    const int s_ = src[e];
    const int d_ = dst[e];
    const float a = ebuf[(long)e * H + h] / fmaxf(ssum[d_ * H + h], 1e-9f);
    atomicAdd(&out[(long)d_ * OS + c], a * Hout[(long)s_ * HD + c]);
}

// logits = out3 @ fc_w + fc_b; sigmoid
__global__ void final_fc(const float* __restrict__ out3, const float* __restrict__ fcw,
                         const float* __restrict__ fcb, float* __restrict__ y) {
    const int n = blockIdx.x * blockDim.x + threadIdx.x;
    if (n >= NNODES) return;
    float acc = fcb[0];
    #pragma unroll
    for (int d = 0; d < 25; ++d) acc += out3[n * 25 + d] * fcw[d];
    y[n] = 1.0f / (1.0f + expf(-acc));
}

extern "C" void kernel_launch(void* const* d_in, const int* in_sizes, int n_in,
                              void* d_out, int out_size, void* d_ws, size_t ws_size,
                              hipStream_t stream) {
    const float* features = (const float*)d_in[0];
    const int*   src      = (const int*)d_in[1];
    const int*   dst      = (const int*)d_in[2];
    const float* W[3]  = {(const float*)d_in[3], (const float*)d_in[7],  (const float*)d_in[11]};
    const float* al[3] = {(const float*)d_in[4], (const float*)d_in[8],  (const float*)d_in[12]};
    const float* ar[3] = {(const float*)d_in[5], (const float*)d_in[9],  (const float*)d_in[13]};
    const float* bi[3] = {(const float*)d_in[6], (const float*)d_in[10], (const float*)d_in[14]};
    const float* fcw = (const float*)d_in[15];
    const float* fcb = (const float*)d_in[16];

    // Workspace layout (floats): ~76 MB total
    float* ws   = (float*)d_ws;
    float* xpad = ws;                                      // N*96 padded features
    float* hbuf = xpad + (size_t)NNODES * 96;              // N*100
    float* xout = hbuf + (size_t)NNODES * 100;             // N*100 (stride varies)
    float* wt   = xout + (size_t)NNODES * 100;             // 112*96 max -> 16384
    float* el   = wt   + (size_t)16384;                    // N*4
    float* er   = el   + (size_t)NNODES * 4;               // N*4
    unsigned* menc = (unsigned*)(er + (size_t)NNODES * 4); // N*4
    float* mf   = (float*)(menc + (size_t)NNODES * 4);     // N*4
    float* ssum = mf   + (size_t)NNODES * 4;               // N*4
    float* ebuf = ssum + (size_t)NNODES * 4;               // E*4

    const int D = 25;
    const int Hh[3]    = {4, 2, 1};
    const int Kk[3]    = {93, 100, 50};
    const int Kpad[3]  = {96, 100, 52};   // K rounded to multiple of 4
    const int HDpad[3] = {112, 64, 32};   // H*D rounded to multiple of 16
    const int OS[3]    = {100, 52, 25};   // output stride = next layer's Kpad

    // Stage padded features once (layers 2/3 inputs are written pre-padded)
    {
        const long tot = (long)NNODES * 96;
        pad_rows<<<(int)((tot + 255) / 256), 256, 0, stream>>>(features, xpad, 93, 96, tot);
    }

    const float* xin = xpad;
    for (int L = 0; L < 3; ++L) {
        const int H = Hh[L], HD = H * D;

        // Stage transposed, padded weights (tiny)
        const int wtot = HDpad[L] * Kpad[L];
        stage_wt<<<(wtot + 255) / 256, 256, 0, stream>>>(W[L], wt, Kk[L], HD,
                                                         Kpad[L], HDpad[L]);

        dim3 g((NNODES / 16 + 7) / 8, HDpad[L] / 16);      // 50000/16 = 3125 tiles
        gemm16x16x4<<<g, 256, 0, stream>>>(xin, wt, hbuf, Kpad[L], HD);

        const int nh = NNODES * H;
        node_scores<<<(nh + 255) / 256, 256, 0, stream>>>(hbuf, al[L], ar[L], el, er, H, D);
        init_nodes<<<(nh + 255) / 256, 256, 0, stream>>>(menc, ssum, nh);

        const int eh = NEDGES * H;
        edge_score_max<<<(eh + 255) / 256, 256, 0, stream>>>(el, er, src, dst, ebuf, menc, H);
        decode_max<<<(nh + 255) / 256, 256, 0, stream>>>(menc, mf, nh);
        edge_expsum<<<(eh + 255) / 256, 256, 0, stream>>>(ebuf, mf, dst, ssum, H);

        const int nos = NNODES * OS[L];
        init_out<<<(nos + 255) / 256, 256, 0, stream>>>(xout, bi[L], HD, OS[L], nos);

        const long ehd = (long)NEDGES * HD;
        edge_aggregate<<<(int)((ehd + 255) / 256), 256, 0, stream>>>(
            ebuf, ssum, hbuf, src, dst, xout, H, D, OS[L]);

        xin = xout;  // layer output (pre-padded stride) feeds next layer's GEMM
    }

    final_fc<<<(NNODES + 255) / 256, 256, 0, stream>>>(xout, fcw, fcb, (float*)d_out);
}